// FastCondGCN_13804024889950
// MI455X (gfx1250) — compile-verified
//
#include <hip/hip_runtime.h>

typedef __attribute__((ext_vector_type(2))) float v2f;
typedef __attribute__((ext_vector_type(8))) float v8f;

#define IN_F  128
#define CTX_F 64
#define HID   128
#define OUT3  384   // 3*HID (x-branch width)
#define OUT2  256   // 2*HID (c-branch width)

__device__ __forceinline__ v8f wmma_f32_k4(v2f a, v2f b, v8f c) {
  // D(16x16,f32) = A(16x4,f32) x B(4x16,f32) + C
  return __builtin_amdgcn_wmma_f32_16x16x4_f32(
      /*neg_a=*/false, a, /*neg_b=*/false, b,
      /*c_mod=*/(short)0, c, /*reuse_a=*/false, /*reuse_b=*/false);
}

// ---------------------------------------------------------------------------
// Tensor Data Mover staging: one tensor_load_to_lds per block pulls a
// (tile_rows x width) f32 tile into LDS, zero-filling rows past tensor_rows.
// D# bit layout per CDNA5 ISA ch.8 (group0: count/lds_addr/global_addr/type,
// group1: data_size/dims/tile dims/stride, groups 2-3 zero for 2-D tensors).
// ---------------------------------------------------------------------------
#if __has_builtin(__builtin_amdgcn_tensor_load_to_lds)
#define USE_TDM 1
typedef __attribute__((ext_vector_type(4))) unsigned tdm_v4u;
typedef __attribute__((ext_vector_type(8))) int      tdm_v8i;
typedef __attribute__((ext_vector_type(4))) int      tdm_v4i;

__device__ __forceinline__ void tdm_load_tile(unsigned lds_off, const float* gptr,
                                              unsigned width, unsigned tile_rows,
                                              unsigned tensor_rows, unsigned row_stride) {
  const unsigned long long ga = (unsigned long long)gptr;
  tdm_v4u g0;
  g0[0] = 1u;                                          // count=1, user mode
  g0[1] = lds_off;                                     // lds_addr (bytes)
  g0[2] = (unsigned)(ga & 0xffffffffull);              // global_addr[31:0]
  g0[3] = (unsigned)((ga >> 32) & 0x01ffffffull)       // global_addr[56:32]
          | (2u << 30);                                // type=2 ("image")
  tdm_v8i g1;
  g1[0] = (int)(2u << 16);                             // data_size=2 (4B), mask=0
  g1[1] = (int)((width & 0xffffu) << 16);              // tensor_dim0[15:0] @ [31:16]
  g1[2] = (int)(((width >> 16) & 0xffffu)              // tensor_dim0[31:16]
          | ((tensor_rows & 0xffffu) << 16));          // tensor_dim1[15:0]
  g1[3] = (int)(((tensor_rows >> 16) & 0xffffu)        // tensor_dim1[31:16]
          | ((width & 0xffffu) << 16));                // tile_dim0
  g1[4] = (int)(tile_rows & 0xffffu);                  // tile_dim1 (tile_dim2=0)
  g1[5] = (int)row_stride;                             // tensor_dim0_stride[31:0]
  g1[6] = 0;                                           // stride hi, dim1_stride
  g1[7] = 0;
  tdm_v4i gz = {};
#if defined(__clang_major__) && __clang_major__ >= 23
  tdm_v8i gz8 = {};
  __builtin_amdgcn_tensor_load_to_lds(g0, g1, gz, gz, gz8, 0);
#else
  __builtin_amdgcn_tensor_load_to_lds(g0, g1, gz, gz, 0);
#endif
#if __has_builtin(__builtin_amdgcn_s_wait_tensorcnt)
  __builtin_amdgcn_s_wait_tensorcnt(0);
#else
  asm volatile("s_wait_tensorcnt 0x0" ::: "memory");
#endif
}
#else
#define USE_TDM 0
#endif

// Stage a (16 x width) f32 tile into LDS. With TDM: wave 0 issues the DMA and
// waits TENSORcnt==0; the workgroup barrier releases the consumers.
template <int WIDTH>
__device__ __forceinline__ void stage_tile(float* lds, const float* g,
                                           int rowBase, int N) {
#if USE_TDM
  if (threadIdx.x < 32) {
    const int rows = (N - rowBase) < 16 ? (N - rowBase) : 16;
    tdm_load_tile((unsigned)(size_t)lds, g + (size_t)rowBase * WIDTH,
                  WIDTH, 16u, (unsigned)rows, WIDTH);
  }
#else
  for (int i = threadIdx.x; i < 16 * WIDTH; i += 256) {
    int r = i / WIDTH, k = i % WIDTH;
    int gr = rowBase + r; if (gr >= N) gr = N - 1;
    lds[i] = g[(size_t)gr * WIDTH + k];
  }
#endif
  __syncthreads();
}

// ex = relu(x @ Wx + bx); cols [0,128) -> agg (seeds segment sum with self_x),
// cols [128,384) -> ws_ex (message columns gathered by edges).
__global__ __launch_bounds__(256)
void gemm_x_kernel(const float* __restrict__ x, const float* __restrict__ Wx,
                   const float* __restrict__ bx, float* __restrict__ agg,
                   float* __restrict__ ws_ex, int N) {
  __shared__ float xt[16 * IN_F];
  const int rowBase = blockIdx.x * 16;
  stage_tile<IN_F>(xt, x, rowBase, N);

  const int wave = threadIdx.x >> 5;
  const int lane = threadIdx.x & 31;
  const int hf   = lane >> 4;   // 0: lanes 0-15, 1: lanes 16-31
  const int lr   = lane & 15;

  for (int t = wave; t < OUT3 / 16; t += 8) {
    const int col = t * 16 + lr;
    v8f acc = {};
    for (int k0 = 0; k0 < IN_F; k0 += 4) {
      const int kk = k0 + 2 * hf;
      v2f a, b;
      a[0] = xt[lr * IN_F + kk];
      a[1] = xt[lr * IN_F + kk + 1];
      b[0] = Wx[(size_t)kk * OUT3 + col];
      b[1] = Wx[(size_t)(kk + 1) * OUT3 + col];
      if (k0 + 8 < IN_F)
        __builtin_prefetch(&Wx[(size_t)(k0 + 8 + 2 * hf) * OUT3 + col], 0, 3);
      acc = wmma_f32_k4(a, b, acc);
    }
    const float bias = bx[col];
    for (int r = 0; r < 8; ++r) {
      const int grow = rowBase + r + 8 * hf;
      if (grow >= N) continue;
      const float v = fmaxf(acc[r] + bias, 0.0f);
      if (col < HID) agg[(size_t)grow * HID + col] = v;
      else           ws_ex[(size_t)grow * OUT2 + (col - HID)] = v;
    }
  }
}

// ec = relu(c @ Wc + bc); cols [0,128) -> c_out directly,
// cols [128,256) -> ws_ec (message columns for edge_cx).
__global__ __launch_bounds__(256)
void gemm_c_kernel(const float* __restrict__ cin, const float* __restrict__ Wc,
                   const float* __restrict__ bc, float* __restrict__ c_out,
                   float* __restrict__ ws_ec, int N) {
  __shared__ float ct[16 * CTX_F];
  const int rowBase = blockIdx.x * 16;
  stage_tile<CTX_F>(ct, cin, rowBase, N);

  const int wave = threadIdx.x >> 5;
  const int lane = threadIdx.x & 31;
  const int hf   = lane >> 4;
  const int lr   = lane & 15;

  for (int t = wave; t < OUT2 / 16; t += 8) {
    const int col = t * 16 + lr;
    v8f acc = {};
    for (int k0 = 0; k0 < CTX_F; k0 += 4) {
      const int kk = k0 + 2 * hf;
      v2f a, b;
      a[0] = ct[lr * CTX_F + kk];
      a[1] = ct[lr * CTX_F + kk + 1];
      b[0] = Wc[(size_t)kk * OUT2 + col];
      b[1] = Wc[(size_t)(kk + 1) * OUT2 + col];
      if (k0 + 8 < CTX_F)
        __builtin_prefetch(&Wc[(size_t)(k0 + 8 + 2 * hf) * OUT2 + col], 0, 3);
      acc = wmma_f32_k4(a, b, acc);
    }
    const float bias = bc[col];
    for (int r = 0; r < 8; ++r) {
      const int grow = rowBase + r + 8 * hf;
      if (grow >= N) continue;
      const float v = fmaxf(acc[r] + bias, 0.0f);
      if (col < HID) c_out[(size_t)grow * HID + col] = v;
      else           ws_ec[(size_t)grow * HID + (col - HID)] = v;
    }
  }
}

// One wave per edge: float4 gather of the 128-float message + atomic scatter
// into agg (global_atomic_add_f32). stride = row stride of the message table.
__global__ __launch_bounds__(256)
void scatter_kernel(const int* __restrict__ src, const int* __restrict__ dst,
                    const float* __restrict__ msg, int stride,
                    float* __restrict__ agg, int E) {
  const int wid  = (int)((blockIdx.x * blockDim.x + threadIdx.x) >> 5);
  const int lane = threadIdx.x & 31;
  if (wid >= E) return;
  const int s = src[wid];
  const int d = dst[wid];
  const float4 v = *(const float4*)(msg + (size_t)s * stride + lane * 4);
  float* out = agg + (size_t)d * HID + lane * 4;
  atomicAdd(out + 0, v.x);
  atomicAdd(out + 1, v.y);
  atomicAdd(out + 2, v.z);
  atomicAdd(out + 3, v.w);
}

// In-place x_out = agg @ Wp + bp over d_out's first half. Each block owns its
// 16 rows exclusively: TDM tile -> LDS -> barrier -> WMMA -> overwrite rows.
__global__ __launch_bounds__(256)
void gemm_p_kernel(float* __restrict__ xio, const float* __restrict__ Wp,
                   const float* __restrict__ bp, int N) {
  __shared__ float at[16 * HID];
  const int rowBase = blockIdx.x * 16;
  stage_tile<HID>(at, xio, rowBase, N);

  const int wave = threadIdx.x >> 5;   // 8 waves -> 8 col tiles of 16
  const int lane = threadIdx.x & 31;
  const int hf   = lane >> 4;
  const int lr   = lane & 15;
  const int col  = wave * 16 + lr;

  v8f acc = {};
  for (int k0 = 0; k0 < HID; k0 += 4) {
    const int kk = k0 + 2 * hf;
    v2f a, b;
    a[0] = at[lr * HID + kk];
    a[1] = at[lr * HID + kk + 1];
    b[0] = Wp[(size_t)kk * HID + col];
    b[1] = Wp[(size_t)(kk + 1) * HID + col];
    if (k0 + 8 < HID)
      __builtin_prefetch(&Wp[(size_t)(k0 + 8 + 2 * hf) * HID + col], 0, 3);
    acc = wmma_f32_k4(a, b, acc);
  }
  const float bias = bp[col];
  for (int r = 0; r < 8; ++r) {
    const int grow = rowBase + r + 8 * hf;
    if (grow >= N) continue;
    xio[(size_t)grow * HID + col] = acc[r] + bias;
  }
}

extern "C" void kernel_launch(void* const* d_in, const int* in_sizes, int n_in,
                              void* d_out, int out_size, void* d_ws, size_t ws_size,
                              hipStream_t stream) {
  const float* x    = (const float*)d_in[0];
  const float* c    = (const float*)d_in[1];
  const int*   exx  = (const int*)d_in[2];   // (2,E): [0..E)=src, [E..2E)=dst
  const int*   exlx = (const int*)d_in[3];   // (2,E): src = row1, dst = row0
  const int*   ecx  = (const int*)d_in[4];   // (2,E): [0..E)=src, [E..2E)=dst
  const float* Wx   = (const float*)d_in[5];
  const float* bx   = (const float*)d_in[6];
  const float* Wc   = (const float*)d_in[7];
  const float* bc   = (const float*)d_in[8];
  const float* Wp   = (const float*)d_in[9];
  const float* bp   = (const float*)d_in[10];

  const int NX = in_sizes[0] / IN_F;
  const int NC = in_sizes[1] / CTX_F;
  const int E  = in_sizes[2] / 2;

  float* x_out = (float*)d_out;                 // doubles as agg accumulator
  float* c_out = x_out + (size_t)NX * HID;
  float* ws_ex = (float*)d_ws;                  // NX x 256 (ex cols H..3H)
  float* ws_ec = ws_ex + (size_t)NX * OUT2;     // NC x 128 (ec cols H..2H)

  dim3 blk(256);
  const int gx = (NX + 15) / 16;
  const int gc = (NC + 15) / 16;
  const int ge = (E + 7) / 8;                   // 8 waves (edges) per block

  // 1) ex GEMM: seeds agg with self_x, emits message columns to workspace.
  gemm_x_kernel<<<gx, blk, 0, stream>>>(x, Wx, bx, x_out, ws_ex, NX);
  // 2) ec GEMM: writes c_out directly, emits message columns to workspace.
  gemm_c_kernel<<<gc, blk, 0, stream>>>(c, Wc, bc, c_out, ws_ec, NC);
  // 3) edge scatters: m_xx (ex cols H:2H), m_xlx (ex cols 2H:3H, reversed
  //    endpoints), m_cx (ec cols H:2H).
  scatter_kernel<<<ge, blk, 0, stream>>>(exx,      exx + E, ws_ex,       OUT2, x_out, E);
  scatter_kernel<<<ge, blk, 0, stream>>>(exlx + E, exlx,    ws_ex + HID, OUT2, x_out, E);
  scatter_kernel<<<ge, blk, 0, stream>>>(ecx,      ecx + E, ws_ec,       HID,  x_out, E);
  // 4) final projection, in place on d_out's x_out half.
  gemm_p_kernel<<<gx, blk, 0, stream>>>(x_out, Wp, bp, NX);
}